// RightsGNN_12309376270615
// MI455X (gfx1250) — compile-verified
//
#include <hip/hip_runtime.h>

typedef float v2f __attribute__((ext_vector_type(2)));
typedef float v8f __attribute__((ext_vector_type(8)));

// =====================================================================
// WMMA GEMM:  Y[N,M] (+)= X[N,K] @ W[M,K]^T (+ bias[M])
// W has row stride ldw (>= K).  Requires N%16==0, M%64==0, K%4==0.
// One wave computes a 16x64 output tile: one A fragment (16x4 f32) is
// reused across 4 V_WMMA_F32_16X16X4_F32 issues against 4 B fragments.
// The K-loop is software-pipelined (register double-buffered) so the
// loads for step k+4 overlap the WMMAs of step k instead of forcing
// s_wait_loadcnt 0 in front of every matrix op.
//
// f32 operand layout (ISA 7.12.2, wave32):
//   A 16x4 : lanes 0-15 row M=lane, {VGPR0,VGPR1}={K+0,K+1};
//            lanes 16-31 same rows, {K+2,K+3}  -> contiguous float2/lane
//   B 4x16 : row striped across lanes in a VGPR (mirror of A)
//   C/D    : VGPR r, lane l -> (row = r + 8*(l>>4), col = l&15)
// =====================================================================
template <int BETA, int HASBIAS>
__global__ __launch_bounds__(128) void gemm64_wmma(
    const float* __restrict__ X, const float* __restrict__ W,
    const float* __restrict__ bias, float* __restrict__ Y,
    int N, int M, int K, int ldw) {
  const int wave = threadIdx.x >> 5;
  const int lane = threadIdx.x & 31;
  const int half = lane >> 4;
  const int lm   = lane & 15;

  const int mt = blockIdx.x;             // 64-wide column tile
  const int nt = blockIdx.y * 4 + wave;  // 16-row tile
  if (nt * 16 >= N) return;              // wave-uniform: EXEC stays all-ones

  const int m0 = mt * 64;
  const float* xp  = X + (size_t)(nt * 16 + lm) * K + 2 * half;
  const float* wp0 = W + (size_t)(m0 +  0 + lm) * ldw + 2 * half;
  const float* wp1 = W + (size_t)(m0 + 16 + lm) * ldw + 2 * half;
  const float* wp2 = W + (size_t)(m0 + 32 + lm) * ldw + 2 * half;
  const float* wp3 = W + (size_t)(m0 + 48 + lm) * ldw + 2 * half;

  v8f c0 = {}, c1 = {}, c2 = {}, c3 = {};

  // prologue: fragments for k = 0
  v2f a  = *(const v2f*)(xp);
  v2f b0 = *(const v2f*)(wp0);
  v2f b1 = *(const v2f*)(wp1);
  v2f b2 = *(const v2f*)(wp2);
  v2f b3 = *(const v2f*)(wp3);

  for (int k = 4; k < K; k += 4) {
    // prefetch the unique X stream ahead (speculative, safe per ISA 10.5)
    __builtin_prefetch(xp + k + 64, 0, 0);
    // issue next step's loads before consuming current fragments
    v2f an  = *(const v2f*)(xp + k);
    v2f bn0 = *(const v2f*)(wp0 + k);
    v2f bn1 = *(const v2f*)(wp1 + k);
    v2f bn2 = *(const v2f*)(wp2 + k);
    v2f bn3 = *(const v2f*)(wp3 + k);
    c0 = __builtin_amdgcn_wmma_f32_16x16x4_f32(false, a, false, b0, (short)0, c0, false, false);
    c1 = __builtin_amdgcn_wmma_f32_16x16x4_f32(false, a, false, b1, (short)0, c1, false, false);
    c2 = __builtin_amdgcn_wmma_f32_16x16x4_f32(false, a, false, b2, (short)0, c2, false, false);
    c3 = __builtin_amdgcn_wmma_f32_16x16x4_f32(false, a, false, b3, (short)0, c3, false, false);
    a = an; b0 = bn0; b1 = bn1; b2 = bn2; b3 = bn3;
  }
  // epilogue K-step
  c0 = __builtin_amdgcn_wmma_f32_16x16x4_f32(false, a, false, b0, (short)0, c0, false, false);
  c1 = __builtin_amdgcn_wmma_f32_16x16x4_f32(false, a, false, b1, (short)0, c1, false, false);
  c2 = __builtin_amdgcn_wmma_f32_16x16x4_f32(false, a, false, b2, (short)0, c2, false, false);
  c3 = __builtin_amdgcn_wmma_f32_16x16x4_f32(false, a, false, b3, (short)0, c3, false, false);

  const int orow = nt * 16 + 8 * half;
  float* yrow = Y + (size_t)orow * M + m0 + lm;

  auto store_tile = [&](const v8f& c, int j) {
    const float bv = HASBIAS ? bias[m0 + j * 16 + lm] : 0.0f;
    float* yp = yrow + j * 16;
#pragma unroll
    for (int r = 0; r < 8; ++r) {
      float v = c[r] + bv;
      if (BETA) v += yp[(size_t)r * M];
      yp[(size_t)r * M] = v;
    }
  };
  store_tile(c0, 0);
  store_tile(c1, 1);
  store_tile(c2, 2);
  store_tile(c3, 3);
}

// ---------------- elementwise helpers ----------------
__global__ void zero_kernel(float* __restrict__ p, size_t n) {
  size_t i = (size_t)blockIdx.x * blockDim.x + threadIdx.x;
  size_t s = (size_t)gridDim.x * blockDim.x;
  for (; i < n; i += s) p[i] = 0.0f;
}

__global__ void relu_kernel(float* __restrict__ p, size_t n) {
  size_t i = (size_t)blockIdx.x * blockDim.x + threadIdx.x;
  size_t s = (size_t)gridDim.x * blockDim.x;
  for (; i < n; i += s) p[i] = fmaxf(p[i], 0.0f);
}

// ---------------- tiny-K projections (Creator K=2, Licensee K=1) ------
__global__ void proj_small_kernel(const float* __restrict__ x,
                                  const float* __restrict__ W,
                                  const float* __restrict__ b,
                                  float* __restrict__ y, int N, int M, int K) {
  size_t t = (size_t)blockIdx.x * blockDim.x + threadIdx.x;
  if (t >= (size_t)N * M) return;
  int n = (int)(t / M), m = (int)(t % M);
  float acc = b[m];
  for (int k = 0; k < K; ++k) acc += x[(size_t)n * K + k] * W[(size_t)m * K + k];
  y[t] = acc;
}

// ---------------- edge aggregation (segment mean) ---------------------
__global__ void edge_count_kernel(const int* __restrict__ edge, int E,
                                  float* __restrict__ cnt) {
  int i = blockIdx.x * blockDim.x + threadIdx.x;
  if (i < E) atomicAdd(&cnt[edge[E + i]], 1.0f);
}

__global__ void scatter_add_kernel(const int* __restrict__ edge, int E,
                                   const float* __restrict__ xsrc,
                                   float* __restrict__ agg, int F) {
  const int chunks = F >> 2;  // float4 chunks per row
  size_t t = (size_t)blockIdx.x * blockDim.x + threadIdx.x;
  if (t >= (size_t)E * chunks) return;
  int e = (int)(t / chunks);
  int c = (int)(t % chunks) << 2;
  int s = edge[e], d = edge[E + e];
  float4 v = *(const float4*)(xsrc + (size_t)s * F + c);
  float* o = agg + (size_t)d * F + c;
  atomicAdd(o + 0, v.x);
  atomicAdd(o + 1, v.y);
  atomicAdd(o + 2, v.z);
  atomicAdd(o + 3, v.w);
}

__global__ void div_count_kernel(float* __restrict__ agg,
                                 const float* __restrict__ cnt,
                                 int Ndst, int F) {
  size_t t = (size_t)blockIdx.x * blockDim.x + threadIdx.x;
  if (t >= (size_t)Ndst * F) return;
  agg[t] *= 1.0f / fmaxf(cnt[t / F], 1.0f);
}

// ---------------- heads ----------------
__global__ void inf_head_kernel(const float* __restrict__ hA2,
                                const int* __restrict__ qidx,
                                const float* __restrict__ W1,
                                const float* __restrict__ b1,
                                const float* __restrict__ W2,
                                const float* __restrict__ b2,
                                float* __restrict__ out, int O) {
  __shared__ float z[128];
  int j = threadIdx.x;
  const float* q = hA2 + (size_t)(*qidx) * O;
  float acc = b1[j];
  const float* wr = W1 + (size_t)j * O;
  for (int k = 0; k < O; ++k) acc += q[k] * wr[k];
  z[j] = fmaxf(acc, 0.0f) * W2[j];
  __syncthreads();
  if (j == 0) {
    float s = b2[0];
    for (int k = 0; k < O; ++k) s += z[k];
    out[0] = s;
  }
}

// qpart[j] = att.b1[j] + sum_k q[k] * att.W1[j, k]   (first O columns of W1)
__global__ void att_qpart_kernel(const float* __restrict__ hA2,
                                 const int* __restrict__ qidx,
                                 const float* __restrict__ W1,
                                 const float* __restrict__ b1,
                                 float* __restrict__ qpart, int O) {
  int j = blockIdx.x * blockDim.x + threadIdx.x;
  if (j >= O) return;
  const float* q = hA2 + (size_t)(*qidx) * O;
  const float* wr = W1 + (size_t)j * (2 * O);
  float acc = b1[j];
  for (int k = 0; k < O; ++k) acc += q[k] * wr[k];
  qpart[j] = acc;
}

// attr[n] = b2 + sum_j relu(Z[n,j]) * W2[j]   (one wave per creator row)
__global__ void att_out_kernel(const float* __restrict__ Z,
                               const float* __restrict__ W2,
                               const float* __restrict__ b2,
                               float* __restrict__ out, int NC, int O) {
  int row = blockIdx.x * (blockDim.x >> 5) + (threadIdx.x >> 5);
  int lane = threadIdx.x & 31;
  if (row >= NC) return;
  float s = 0.0f;
  for (int k = lane; k < O; k += 32) s += fmaxf(Z[(size_t)row * O + k], 0.0f) * W2[k];
  for (int off = 16; off > 0; off >>= 1) s += __shfl_xor(s, off);
  if (lane == 0) out[row] = s + b2[0];
}

// =====================================================================
// Host orchestration
// =====================================================================
static void launch_gemm(int beta, int hasbias, const float* X, const float* W,
                        const float* bias, float* Y, int N, int M, int K,
                        int ldw, hipStream_t st) {
  dim3 grid(M / 64, (unsigned)((N + 63) / 64));
  if (beta) {
    if (hasbias) gemm64_wmma<1, 1><<<grid, 128, 0, st>>>(X, W, bias, Y, N, M, K, ldw);
    else         gemm64_wmma<1, 0><<<grid, 128, 0, st>>>(X, W, bias, Y, N, M, K, ldw);
  } else {
    if (hasbias) gemm64_wmma<0, 1><<<grid, 128, 0, st>>>(X, W, bias, Y, N, M, K, ldw);
    else         gemm64_wmma<0, 0><<<grid, 128, 0, st>>>(X, W, bias, Y, N, M, K, ldw);
  }
}

static void launch_zero(float* p, size_t n, hipStream_t st) {
  unsigned blocks = (unsigned)((n + 255) / 256);
  zero_kernel<<<blocks, 256, 0, st>>>(p, n);
}

extern "C" void kernel_launch(void* const* d_in, const int* in_sizes, int n_in,
                              void* d_out, int out_size, void* d_ws,
                              size_t ws_size, hipStream_t stream) {
  constexpr int NA = 50000, NC = 20000, NL = 10000, H = 256, O = 128, FA = 512;

  // ---- top-level inputs (setup_inputs() insertion order) ----
  const float* x_asset    = (const float*)d_in[0];
  const float* x_creator  = (const float*)d_in[1];
  const float* x_licensee = (const float*)d_in[2];
  const int* edges[6] = {(const int*)d_in[3], (const int*)d_in[4],
                         (const int*)d_in[5], (const int*)d_in[6],
                         (const int*)d_in[7], (const int*)d_in[8]};
  int E[6];
  for (int i = 0; i < 6; ++i) E[i] = in_sizes[3 + i] / 2;
  const int* qidx = (const int*)d_in[9];

  // ---- params: jax pytree flatten (sorted dict keys), leaves from 10 ----
  // att(10..13): W1[128,256] W2[128] b1[128] b2[1]
  // conv1(14..31), conv2(32..49): per sorted rel name {Wl, Wr, bl}
  // inf(50..53): W1[128,128] W2[128] b1[128] b2[1]
  // proj(54..59): Asset{W[256,512],b} Creator{W[256,2],b} Licensee{W[256,1],b}
  const float* attW1 = (const float*)d_in[10];
  const float* attW2 = (const float*)d_in[11];
  const float* attb1 = (const float*)d_in[12];
  const float* attb2 = (const float*)d_in[13];
  const float* infW1 = (const float*)d_in[50];
  const float* infW2 = (const float*)d_in[51];
  const float* infb1 = (const float*)d_in[52];
  const float* infb2 = (const float*)d_in[53];
  const float* projAW = (const float*)d_in[54];
  const float* projAb = (const float*)d_in[55];
  const float* projCW = (const float*)d_in[56];
  const float* projCb = (const float*)d_in[57];
  const float* projLW = (const float*)d_in[58];
  const float* projLb = (const float*)d_in[59];

  // relation table: processing order matches edges[] above
  // name          : src dst conv1-leaf conv2-leaf
  // created_by    :  A   C     14         32
  // licensed_to   :  A   L     20         38
  // similar_to    :  A   A     29         47
  // flagged_with  :  A   A     17         35
  // rev_created_by:  C   A     23         41
  // rev_licensed_to: L  A     26         44
  const int srcT[6] = {0, 0, 0, 0, 1, 2};
  const int dstT[6] = {1, 2, 0, 0, 0, 0};
  const int p1[6] = {14, 20, 29, 17, 23, 26};
  const int p2[6] = {32, 38, 47, 35, 41, 44};

  // ---- workspace layout (floats) ----
  float* ws = (float*)d_ws;
  size_t off = 0;
  float* hA0 = ws + off; off += (size_t)NA * H;
  float* hC0 = ws + off; off += (size_t)NC * H;
  float* hL0 = ws + off; off += (size_t)NL * H;
  float* hA1 = ws + off; off += (size_t)NA * H;
  float* hC1 = ws + off; off += (size_t)NC * H;
  float* hL1 = ws + off; off += (size_t)NL * H;
  float* agg = ws + off; off += (size_t)NA * H;   // reused per relation
  float* cnt = ws + off; off += (size_t)NA;
  float* Zat = ws + off; off += (size_t)NC * O;
  float* qpart = ws + off; off += 128;

  // ---- output regions ----
  float* out_inf  = (float*)d_out;
  float* out_attr = out_inf + 1;
  float* out_hA   = out_attr + NC;
  float* out_hC   = out_hA + (size_t)NA * O;
  float* out_hL   = out_hC + (size_t)NC * O;

  const int Ndst[3] = {NA, NC, NL};

  // ================= 1) node-type projections =================
  launch_gemm(0, 1, x_asset, projAW, projAb, hA0, NA, H, FA, FA, stream);
  {
    size_t tC = (size_t)NC * H, tL = (size_t)NL * H;
    proj_small_kernel<<<(unsigned)((tC + 255) / 256), 256, 0, stream>>>(
        x_creator, projCW, projCb, hC0, NC, H, 2);
    proj_small_kernel<<<(unsigned)((tL + 255) / 256), 256, 0, stream>>>(
        x_licensee, projLW, projLb, hL0, NL, H, 1);
  }

  // ================= 2) two hetero SAGE layers =================
  for (int layer = 0; layer < 2; ++layer) {
    const float* hin[3];
    float* hout[3];
    int Mout;
    if (layer == 0) {
      hin[0] = hA0; hin[1] = hC0; hin[2] = hL0;
      hout[0] = hA1; hout[1] = hC1; hout[2] = hL1;
      Mout = H;   // 256
    } else {
      hin[0] = hA1; hin[1] = hC1; hin[2] = hL1;
      hout[0] = out_hA; hout[1] = out_hC; hout[2] = out_hL;
      Mout = O;   // 128
    }
    bool first[3] = {true, true, true};

    for (int r = 0; r < 6; ++r) {
      const int d = dstT[r], s = srcT[r];
      const int nd = Ndst[d];
      const int pl = (layer == 0) ? p1[r] : p2[r];
      const float* Wl = (const float*)d_in[pl + 0];
      const float* Wr = (const float*)d_in[pl + 1];
      const float* bl = (const float*)d_in[pl + 2];

      // mean-aggregate src features at dst
      launch_zero(agg, (size_t)nd * H, stream);
      launch_zero(cnt, (size_t)nd, stream);
      edge_count_kernel<<<(E[r] + 255) / 256, 256, 0, stream>>>(edges[r], E[r], cnt);
      {
        size_t tot = (size_t)E[r] * (H >> 2);
        scatter_add_kernel<<<(unsigned)((tot + 255) / 256), 256, 0, stream>>>(
            edges[r], E[r], hin[s], agg, H);
      }
      {
        size_t tot = (size_t)nd * H;
        div_count_kernel<<<(unsigned)((tot + 255) / 256), 256, 0, stream>>>(
            agg, cnt, nd, H);
      }

      // out[d] (+)= lin_l(mean) + bl ;  then += lin_r(h[d])
      launch_gemm(first[d] ? 0 : 1, 1, agg, Wl, bl, hout[d], nd, Mout, H, H, stream);
      first[d] = false;
      launch_gemm(1, 0, hin[d], Wr, nullptr, hout[d], nd, Mout, H, H, stream);
    }

    for (int t = 0; t < 3; ++t) {
      size_t tot = (size_t)Ndst[t] * Mout;
      relu_kernel<<<(unsigned)((tot + 255) / 256), 256, 0, stream>>>(hout[t], tot);
    }
  }

  // ================= 3) heads =================
  inf_head_kernel<<<1, 128, 0, stream>>>(out_hA, qidx, infW1, infb1, infW2,
                                         infb2, out_inf, O);

  // attribution: pair = [q, hC];  z2 = relu(pair @ W1^T + b1)
  //   q-part + b1 folds into a per-column bias; hC part is a WMMA GEMM.
  att_qpart_kernel<<<1, 128, 0, stream>>>(out_hA, qidx, attW1, attb1, qpart, O);
  launch_gemm(0, 1, out_hC, attW1 + O /* second half of rows */, qpart, Zat,
              NC, O, O, 2 * O, stream);
  att_out_kernel<<<(NC + 3) / 4, 128, 0, stream>>>(Zat, attW2, attb2, out_attr,
                                                   NC, O);

  (void)n_in; (void)out_size; (void)ws_size;
}